// MonoSIMTorch_45784351375917
// MI455X (gfx1250) — compile-verified
//
#include <hip/hip_runtime.h>
#include <stdint.h>

typedef float v2f __attribute__((ext_vector_type(2)));
typedef float v8f __attribute__((ext_vector_type(8)));
typedef unsigned int u32x4 __attribute__((ext_vector_type(4)));
typedef int i32x4 __attribute__((ext_vector_type(4)));
typedef int i32x8 __attribute__((ext_vector_type(8)));

#define DON_TWO_PI 6.28318530717958647692f

#if defined(__has_builtin)
#if __has_builtin(__builtin_amdgcn_tensor_load_to_lds) && \
    __has_builtin(__builtin_amdgcn_s_wait_tensorcnt)
#define DON_USE_TDM 1
#endif
#endif

// ---------------------------------------------------------------------------
// y[l][m] = exp(i * 2*pi * sigmoid(p[l][m]))  -> interleaved complex float2
// ---------------------------------------------------------------------------
__global__ void don_phase_kernel(const float* __restrict__ p,
                                 float* __restrict__ y, int n) {
    int i = blockIdx.x * blockDim.x + threadIdx.x;
    if (i < n) {
        float s  = 1.0f / (1.0f + __expf(-p[i]));
        float th = DON_TWO_PI * s;
        float sn, cs;
        __sincosf(th, &sn, &cs);
        y[2 * i]     = cs;   // real
        y[2 * i + 1] = sn;   // imag
    }
}

#ifdef DON_USE_TDM
// ---------------------------------------------------------------------------
// Issue one TDM 2D tile load: global (row-major, 8B complex elements) -> LDS,
// with LDS row padding applied by the TDM pad engine.
// Descriptor packing per CDNA5 ISA 8.3/8.4 (D# groups 0 and 1); groups 2/3
// zero (2D tensor). 6-arg builtin form: (g0, g1, g2, g3, g_extra, cpol).
// ---------------------------------------------------------------------------
__device__ __forceinline__ void don_tdm_load_2d(
    uint32_t lds_byte_off, const void* gptr,
    uint32_t tensor_d0, uint32_t tensor_d1, uint64_t stride0_elems,
    uint32_t tile_d0, uint32_t tile_d1,
    uint32_t pad_interval_code, uint32_t pad_amount_code)
{
    uint64_t ga = (uint64_t)(uintptr_t)gptr;
    u32x4 g0;
    g0.x = 1u;                                            // count=1 (valid D#)
    g0.y = lds_byte_off;                                  // [63:32]  lds_addr
    g0.z = (uint32_t)(ga & 0xFFFFFFFFu);                  // [95:64]  gaddr lo
    g0.w = (uint32_t)((ga >> 32) & 0x1FFFFFFu)            // [120:96] gaddr hi
         | (2u << 30);                                    // [127:126] type=2
    i32x8 g1;
    g1[0] = (int)((3u << 16)                              // data_size = 8B
                | (1u << 20)                              // pad_enable
                | (pad_interval_code << 22)
                | (pad_amount_code << 25));
    g1[1] = (int)((tensor_d0 & 0xFFFFu) << 16);           // [63:48] dim0 lo
    g1[2] = (int)((tensor_d0 >> 16) |
                  ((tensor_d1 & 0xFFFFu) << 16));         // dim0 hi | dim1 lo
    g1[3] = (int)((tensor_d1 >> 16) | (tile_d0 << 16));   // dim1 hi | tile0
    g1[4] = (int)(tile_d1 & 0xFFFFu);                     // tile1 | tile2=0
    g1[5] = (int)(uint32_t)(stride0_elems & 0xFFFFFFFFu); // stride0 lo
    g1[6] = (int)(uint32_t)((stride0_elems >> 32) & 0xFFFFu); // stride0 hi
    g1[7] = 0;                                            // stride1 = 0
    i32x4 z4 = {0, 0, 0, 0};
    i32x8 z8 = {0, 0, 0, 0, 0, 0, 0, 0};
    __builtin_amdgcn_tensor_load_to_lds(g0, g1, z4, z4, z8, 0);
}
#endif

// ---------------------------------------------------------------------------
// One 16-deep K panel of complex WMMAs (4 k-steps of K=4).
// ---------------------------------------------------------------------------
__device__ __forceinline__ void don_compute_panel(
    const float2 (*pA)[18], const float2 (*pB)[66],
    int waveM, int waveN, int lm, int lk,
    v8f accr[2][2], v8f acci[2][2])
{
    #pragma unroll
    for (int kk = 0; kk < 4; ++kk) {
        const int k0 = kk * 4 + lk;

        v2f ar[2], ai[2], ain[2];
        #pragma unroll
        for (int ms = 0; ms < 2; ++ms) {
            int m = waveM + ms * 16 + lm;
            float2 a0 = pA[m][k0];
            float2 a1 = pA[m][k0 + 1];
            ar[ms].x = a0.x; ar[ms].y = a1.x;
            ai[ms].x = a0.y; ai[ms].y = a1.y;
            ain[ms] = -ai[ms];   // F32 WMMA has no A-negate -> VALU neg
        }
        v2f br[2], bi[2];
        #pragma unroll
        for (int ns = 0; ns < 2; ++ns) {
            int n = waveN + ns * 16 + lm;
            float2 b0 = pB[k0][n];
            float2 b1 = pB[k0 + 1][n];
            br[ns].x = b0.x; br[ns].y = b1.x;
            bi[ns].x = b0.y; bi[ns].y = b1.y;
        }

        #pragma unroll
        for (int ms = 0; ms < 2; ++ms) {
            #pragma unroll
            for (int ns = 0; ns < 2; ++ns) {
                // Cr += Ar*Br - Ai*Bi ; Ci += Ar*Bi + Ai*Br
                accr[ms][ns] = __builtin_amdgcn_wmma_f32_16x16x4_f32(
                    false, ar[ms], false, br[ns], (short)0, accr[ms][ns], false, false);
                accr[ms][ns] = __builtin_amdgcn_wmma_f32_16x16x4_f32(
                    false, ain[ms], false, bi[ns], (short)0, accr[ms][ns], false, false);
                acci[ms][ns] = __builtin_amdgcn_wmma_f32_16x16x4_f32(
                    false, ar[ms], false, bi[ns], (short)0, acci[ms][ns], false, false);
                acci[ms][ns] = __builtin_amdgcn_wmma_f32_16x16x4_f32(
                    false, ai[ms], false, br[ns], (short)0, acci[ms][ns], false, false);
            }
        }
    }
}

// ---------------------------------------------------------------------------
// Complex GEMM with optional row scale:
//   C[M x N] = diag(yrow) * (A[M x K] @ B[K x N])   (complex64, row-major,
//   interleaved re/im).  yrow == nullptr -> no scaling.
//
// Block: 128 threads = 4 wave32s; block tile 64x64; K panel 16, double-
// buffered in LDS. Panels are fetched by the Tensor Data Mover (wave 0
// issues, TENSORcnt-tracked) and overlapped with WMMA compute.
// ---------------------------------------------------------------------------
__global__ __launch_bounds__(128)
void don_cgemm_scale(const float* __restrict__ A,
                     const float* __restrict__ Bm,
                     float* __restrict__ C,
                     const float* __restrict__ yrow,
                     int M, int N, int K) {
    // rows padded by 2 complex elements (TDM pad: A 32dw->code4, B 128dw->code6,
    // pad 4 dwords -> code 3). A frag reads: float addr stride 36 -> conflict-free.
    __shared__ float2 sA[2][64][18];
    __shared__ float2 sB[2][16][66];

    const int t    = threadIdx.x;
    const int lane = t & 31;
    const int wave = t >> 5;
    const int lm   = lane & 15;
    const int lhi  = lane >> 4;
    const int lk   = lhi * 2;

    const int tileM = blockIdx.y * 64;
    const int tileN = blockIdx.x * 64;
    const int waveM = (wave >> 1) * 32;
    const int waveN = (wave & 1) * 32;

    const float2* A2 = (const float2*)A;
    const float2* B2 = (const float2*)Bm;

    v8f accr[2][2] = {};
    v8f acci[2][2] = {};

    const int kPanels = K >> 4;

#ifdef DON_USE_TDM
    const uint32_t ldsA[2] = {(uint32_t)(uintptr_t)&sA[0][0][0],
                              (uint32_t)(uintptr_t)&sA[1][0][0]};
    const uint32_t ldsB[2] = {(uint32_t)(uintptr_t)&sB[0][0][0],
                              (uint32_t)(uintptr_t)&sB[1][0][0]};

    if (wave == 0) {   // prefetch panel 0 into buffer 0 (EXEC-independent DMA)
        don_tdm_load_2d(ldsA[0], &A2[(size_t)tileM * K],
                        (uint32_t)K, (uint32_t)M, (uint64_t)K, 16, 64, 4, 3);
        don_tdm_load_2d(ldsB[0], &B2[tileN],
                        (uint32_t)N, (uint32_t)K, (uint64_t)N, 64, 16, 6, 3);
    }
    for (int kt = 0; kt < kPanels; ++kt) {
        const int buf = kt & 1;
        if (wave == 0) {
            if (kt + 1 < kPanels) {
                don_tdm_load_2d(ldsA[buf ^ 1],
                                &A2[(size_t)tileM * K + (size_t)(kt + 1) * 16],
                                (uint32_t)K, (uint32_t)M, (uint64_t)K, 16, 64, 4, 3);
                don_tdm_load_2d(ldsB[buf ^ 1],
                                &B2[(size_t)(kt + 1) * 16 * N + tileN],
                                (uint32_t)N, (uint32_t)K, (uint64_t)N, 64, 16, 6, 3);
                // TDM ops complete in order per wave: <=2 outstanding means
                // the current panel's two loads have landed.
                __builtin_amdgcn_s_wait_tensorcnt((short)2);
            } else {
                __builtin_amdgcn_s_wait_tensorcnt((short)0);
            }
        }
        __syncthreads();   // publish panel `buf` to all waves
        don_compute_panel(sA[buf], sB[buf], waveM, waveN, lm, lk, accr, acci);
        __syncthreads();   // all reads of `buf^1`'s victim done before next DMA
    }
#else
    // Fallback: cooperative float4 staging into the same padded layout.
    for (int kt = 0; kt < kPanels; ++kt) {
        const int buf = kt & 1;
        __syncthreads();
        #pragma unroll
        for (int j = 0; j < 4; ++j) {          // A: 512 float4 / 128 threads
            int idx = t + 128 * j;
            int r = idx >> 3, c = (idx & 7) * 2;
            *(float4*)&sA[buf][r][c] =
                *(const float4*)&A2[(size_t)(tileM + r) * K + kt * 16 + c];
        }
        #pragma unroll
        for (int j = 0; j < 4; ++j) {          // B: 512 float4 / 128 threads
            int idx = t + 128 * j;
            int r = idx >> 5, c = (idx & 31) * 2;
            *(float4*)&sB[buf][r][c] =
                *(const float4*)&B2[(size_t)(kt * 16 + r) * N + tileN + c];
        }
        __syncthreads();
        don_compute_panel(sA[buf], sB[buf], waveM, waveN, lm, lk, accr, acci);
    }
#endif

    // ---- epilogue: optional per-row complex scale, store interleaved ----
    float2* C2 = (float2*)C;
    #pragma unroll
    for (int ms = 0; ms < 2; ++ms) {
        #pragma unroll
        for (int ns = 0; ns < 2; ++ns) {
            #pragma unroll
            for (int r = 0; r < 8; ++r) {
                int m = tileM + waveM + ms * 16 + lhi * 8 + r;  // vgpr r -> M=r / M=r+8
                int n = tileN + waveN + ns * 16 + lm;
                float cr = accr[ms][ns][r];
                float ci = acci[ms][ns][r];
                float tr, ti;
                if (yrow != nullptr) {
                    float yr = yrow[2 * m];
                    float yi = yrow[2 * m + 1];
                    tr = yr * cr - yi * ci;
                    ti = yr * ci + yi * cr;
                } else {
                    tr = cr;
                    ti = ci;
                }
                C2[(size_t)m * N + n] = make_float2(tr, ti);
            }
        }
    }
}

// ---------------------------------------------------------------------------
// Launch: out = WL @ (Y7 W7 ... Y1 W1 Y0 W0 x)   (right-to-left propagation,
// ~138 GFLOP vs ~262 GFLOP for the reference's left-to-right order)
// ---------------------------------------------------------------------------
extern "C" void kernel_launch(void* const* d_in, const int* in_sizes, int n_in,
                              void* d_out, int out_size, void* d_ws, size_t ws_size,
                              hipStream_t stream) {
    (void)in_sizes; (void)n_in; (void)out_size; (void)ws_size;
    constexpr int L    = 8;
    constexpr int M    = 2048;
    constexpr int NIN  = 1024;
    constexpr int NOUT = 1024;
    constexpr int Bc   = 512;

    const float* x    = (const float*)d_in[0];  // [NIN  x Bc ] complex
    const float* W0   = (const float*)d_in[1];  // [M    x NIN] complex
    const float* WL   = (const float*)d_in[2];  // [NOUT x M  ] complex
    const float* Wint = (const float*)d_in[3];  // [L-1, M, M ] complex
    const float* pp   = (const float*)d_in[4];  // [L, M] float32
    float* out = (float*)d_out;                 // [NOUT x Bc] complex interleaved

    // workspace layout
    float* y  = (float*)d_ws;                                          // L*M complex
    float* T0 = (float*)((char*)d_ws + (1 << 18));                     // M*Bc complex
    float* T1 = (float*)((char*)d_ws + (1 << 18) +
                         (size_t)M * Bc * 2 * sizeof(float));

    // 1) y = exp(i * 2pi * sigmoid(p))
    don_phase_kernel<<<(L * M + 255) / 256, 256, 0, stream>>>(pp, y, L * M);

    dim3 blk(128);

    // 2) T0 = diag(y0) * (W0 @ x)        M=2048, K=1024, N=512
    don_cgemm_scale<<<dim3(Bc / 64, M / 64), blk, 0, stream>>>(
        W0, x, T0, y, M, Bc, NIN);

    // 3) T = diag(y_l) * (W_int[l-1] @ T), ping-pong
    for (int l = 1; l < L; ++l) {
        const float* Ain = Wint + (size_t)(l - 1) * M * M * 2;
        const float* src = (l & 1) ? T0 : T1;
        float*       dst = (l & 1) ? T1 : T0;
        don_cgemm_scale<<<dim3(Bc / 64, M / 64), blk, 0, stream>>>(
            Ain, src, dst, y + (size_t)l * M * 2, M, Bc, M);
    }

    // 4) out = WL @ T7   (l=7 wrote T1)   M=1024, K=2048, N=512, no scale
    don_cgemm_scale<<<dim3(Bc / 64, NOUT / 64), blk, 0, stream>>>(
        WL, T1, out, nullptr, NOUT, Bc, M);
}